// FastVideoCompressor_65077344469075
// MI455X (gfx1250) — compile-verified
//
#include <hip/hip_runtime.h>
#include <hip/hip_bf16.h>
#include <math.h>

// ---------------------------------------------------------------------------
// FastVideoCompressor on MI455X (gfx1250)
//
// The reference scan is sequential in appearance only: the carry is replaced
// by the *current* frame whenever is_kf fires, and is_kf depends only on the
// current key.  So:
//   phase 1: selector GEMM over all t -> is_kf flags        (parallel)
//   phase 2: O(S) scalar scan -> index of last keyframe < t (trivial)
//   phase 3: gather prev_k/prev_v, then 8 big fp32 GEMMs    (parallel)
// All GEMMs use V_WMMA_F32_16X16X4_F32 (fp32-exact matrix path on CDNA5).
//
// This revision interleaves the W tile in LDS by K-pairs so each B-fragment
// is a single aligned ds_load_b64 (no v_mov operand patching in the WMMA
// inner loop).
// ---------------------------------------------------------------------------

typedef float v2f __attribute__((ext_vector_type(2)));
typedef float v8f __attribute__((ext_vector_type(8)));

#define HS   2048
#define CS   1024
#define SEQ  4096

// ----------------------------- WMMA GEMM -----------------------------------
// C[M,N] = epilogue( X[M,K] @ W[K,N] )
//   X optionally split at column K0 (cols [0,K0) from X0 stride K0,
//   cols [K0,K) from X1 stride K-K0).  epilogue: +bias, opt ReLU, opt +add.
// Block tile 128x64, 8 wave32 waves (4 M x 2 N), each wave 32x32 (2x2 WMMA).
// K staged in double-buffered LDS chunks of 32.

#define BM 128
#define BN 64
#define KT 32
#define XSTR 36    // padded X row stride: float4-aligned stores, conflict-free b64 reads
#define WSTR2 160  // padded W pair-row stride (floats): (WSTR2/2)%32==16 -> half-waves
                   // hit disjoint bank halves on B-fragment b64 reads

__global__ __launch_bounds__(256)
void gemm_f32_wmma(const float* __restrict__ X0, const float* __restrict__ X1,
                   const float* __restrict__ W,  const float* __restrict__ bias,
                   const float* __restrict__ add, float* __restrict__ out,
                   int M, int N, int K, int K0, int relu)
{
    __shared__ float sX[2][BM * XSTR];              // 36864 B
    __shared__ float sW[2][(KT / 2) * WSTR2];       // 20480 B  (total 57344 B)

    const int tid   = threadIdx.x;
    const int bm    = blockIdx.y * BM;
    const int bn    = blockIdx.x * BN;
    const int lane  = tid & 31;
    const int wid   = tid >> 5;
    const int waveM = wid & 3;          // 0..3
    const int waveN = wid >> 2;         // 0..1
    const int lo    = lane & 15;
    const int hi    = lane >> 4;        // half-wave selects K pair / row half

    v8f acc[2][2] = {};

    // X staging: 128 rows x 32 cols, 4 float4 per thread
    const int xtrow = tid >> 3;            // 0..31 (x4 row passes)
    const int xtcol = (tid & 7) * 4;       // 0..28
    // W staging: 32 rows x 64 cols; each thread handles a (row-pair, col-pair):
    //   load float2 from row k and row k+1 at col wtc, emit one interleaved float4.
    const int wtc = (tid & 31) * 2;        // col pair base 0..62
    const int wtk = (tid >> 5) * 2;        // row pair base 0..14 (x2 passes)
    const int K1 = K - K0;

    float4 xr[4];
    float4 wr[2];

    auto loadX = [&](int kc) {
        const int gcol = kc + xtcol;
#pragma unroll
        for (int rr = 0; rr < 4; ++rr) {
            const int row = bm + xtrow + 32 * rr;
            const float* p = (gcol < K0)
                ? (X0 + (size_t)row * K0 + gcol)
                : (X1 + (size_t)row * K1 + (gcol - K0));
            xr[rr] = *(const float4*)p;
        }
    };
    auto loadW = [&](int kc) {
#pragma unroll
        for (int rr = 0; rr < 2; ++rr) {
            const int k = kc + wtk + 16 * rr;
            v2f w0 = *(const v2f*)(W + (size_t)k * N + bn + wtc);
            v2f w1 = *(const v2f*)(W + (size_t)(k + 1) * N + bn + wtc);
            wr[rr] = make_float4(w0.x, w1.x, w0.y, w1.y);  // [c,k] [c,k+1] [c+1,k] [c+1,k+1]
        }
    };
    auto stageX = [&](int b) {
#pragma unroll
        for (int rr = 0; rr < 4; ++rr)
            *(float4*)&sX[b][(xtrow + 32 * rr) * XSTR + xtcol] = xr[rr];
    };
    auto stageW = [&](int b) {
#pragma unroll
        for (int rr = 0; rr < 2; ++rr)
            *(float4*)&sW[b][(wtk / 2 + 8 * rr) * WSTR2 + wtc * 2] = wr[rr];
    };

    auto compute = [&](int b) {
        const int c0 = (waveN * 32 + lo) * 2;       // interleaved column offset
        const int r0 = (waveM * 32 + lo) * XSTR;
        const int r1 = (waveM * 32 + 16 + lo) * XSTR;
#pragma unroll
        for (int kk = 0; kk < KT; kk += 4) {
            const int ka = kk + 2 * hi;             // even -> 8B-aligned A pair
            const int kw = (kk / 2 + hi) * WSTR2;   // interleaved B pair row
            v2f a0 = *(const v2f*)&sX[b][r0 + ka];
            v2f a1 = *(const v2f*)&sX[b][r1 + ka];
            v2f b0 = *(const v2f*)&sW[b][kw + c0];
            v2f b1 = *(const v2f*)&sW[b][kw + c0 + 32];
            acc[0][0] = __builtin_amdgcn_wmma_f32_16x16x4_f32(false, a0, false, b0,
                                                              (short)0, acc[0][0], false, false);
            acc[0][1] = __builtin_amdgcn_wmma_f32_16x16x4_f32(false, a0, false, b1,
                                                              (short)0, acc[0][1], false, false);
            acc[1][0] = __builtin_amdgcn_wmma_f32_16x16x4_f32(false, a1, false, b0,
                                                              (short)0, acc[1][0], false, false);
            acc[1][1] = __builtin_amdgcn_wmma_f32_16x16x4_f32(false, a1, false, b1,
                                                              (short)0, acc[1][1], false, false);
        }
    };

    const int nc = K / KT;
    loadX(0); loadW(0);
    stageX(0); stageW(0);
    __syncthreads();

    for (int c = 0; c < nc; ++c) {
        const int cb = c & 1;
        if (c + 1 < nc) { loadX((c + 1) * KT); loadW((c + 1) * KT); }
        compute(cb);
        if (c + 1 < nc) {
            stageX(cb ^ 1); stageW(cb ^ 1);
            __syncthreads();
        }
    }

    // epilogue:  v = acc + bias[n];  if relu v=max(v,0);  if add v += add[m,n]
#pragma unroll
    for (int ti = 0; ti < 2; ++ti) {
#pragma unroll
        for (int tj = 0; tj < 2; ++tj) {
            const int col = bn + waveN * 32 + 16 * tj + lo;
            const float bv = bias ? bias[col] : 0.0f;
#pragma unroll
            for (int r = 0; r < 8; ++r) {
                const int row = bm + waveM * 32 + 16 * ti + r + 8 * hi;
                float v = acc[ti][tj][r] + bv;
                if (relu) v = fmaxf(v, 0.0f);
                if (add)  v += add[(size_t)row * N + col];
                out[(size_t)row * N + col] = v;
            }
        }
    }
}

// --------------------------- small kernels ---------------------------------

// score[t] = sigmoid( hid[t,:] . W2 + b2 );  flag[t] = interval_kf || score>0.2
__global__ __launch_bounds__(256)
void sel_score_kernel(const float* __restrict__ hid, const float* __restrict__ W2,
                      const float* __restrict__ b2, const int* __restrict__ frame_count,
                      int* __restrict__ flags, int Hh)
{
    const int t = blockIdx.x;
    float s = 0.0f;
    for (int i = threadIdx.x; i < Hh; i += 256)
        s += hid[(size_t)t * Hh + i] * W2[i];
    __shared__ float red[256];
    red[threadIdx.x] = s;
    __syncthreads();
    for (int off = 128; off > 0; off >>= 1) {
        if (threadIdx.x < off) red[threadIdx.x] += red[threadIdx.x + off];
        __syncthreads();
    }
    if (threadIdx.x == 0) {
        const float sc = 1.0f / (1.0f + __expf(-(red[0] + b2[0])));
        const int interval = (frame_count[0] % 8) == 0;
        flags[t] = (interval || (sc > 0.2f)) ? 1 : 0;
    }
}

// src[t] = index of last keyframe strictly before t, else -1
__global__ void kf_scan_kernel(const int* __restrict__ flags, int* __restrict__ src, int S)
{
    if (blockIdx.x == 0 && threadIdx.x == 0) {
        int last = -1;
        for (int t = 0; t < S; ++t) {
            src[t] = last;
            if (flags[t]) last = t;
        }
    }
}

// prev[t,h] = (src[t] >= 0) ? base[src[t], h] : 0
__global__ __launch_bounds__(256)
void gather_prev_kernel(const float* __restrict__ base, const int* __restrict__ src,
                        float* __restrict__ prev, int H)
{
    const int t = blockIdx.y;
    const int h = blockIdx.x * 256 + threadIdx.x;
    const int s = src[t];
    prev[(size_t)t * H + h] = (s >= 0) ? base[(size_t)s * H + h] : 0.0f;
}

// out = a - b (elementwise)
__global__ __launch_bounds__(256)
void sub_kernel(const float* __restrict__ a, const float* __restrict__ b,
                float* __restrict__ out, size_t n)
{
    const size_t i = (size_t)blockIdx.x * 256 + threadIdx.x;
    if (i < n) out[i] = a[i] - b[i];
}

// pos = flag ? cur : rec;  blend = sigmoid(5*imp);  out = blend*cur + (1-blend)*pos
__global__ __launch_bounds__(256)
void final_kernel(const float* __restrict__ cur, const float* __restrict__ rec,
                  const float* __restrict__ imp, const int* __restrict__ flags,
                  float* __restrict__ out, int H)
{
    const int t = blockIdx.y;
    const int h = blockIdx.x * 256 + threadIdx.x;
    const size_t idx = (size_t)t * H + h;
    const float ck  = cur[idx];
    const float pos = flags[t] ? ck : rec[idx];
    const float bl  = 1.0f / (1.0f + __expf(-imp[t] * 5.0f));
    out[idx] = bl * ck + (1.0f - bl) * pos;
}

// ------------------------------- launch ------------------------------------

extern "C" void kernel_launch(void* const* d_in, const int* in_sizes, int n_in,
                              void* d_out, int out_size, void* d_ws, size_t ws_size,
                              hipStream_t stream)
{
    const float* keys   = (const float*)d_in[0];
    const float* values = (const float*)d_in[1];
    const float* imp    = (const float*)d_in[2];
    const float* km_W1  = (const float*)d_in[3];
    const float* km_b1  = (const float*)d_in[4];
    const float* km_W2  = (const float*)d_in[5];
    const float* km_b2  = (const float*)d_in[6];
    const float* vm_W1  = (const float*)d_in[7];
    const float* vm_b1  = (const float*)d_in[8];
    const float* vm_W2  = (const float*)d_in[9];
    const float* vm_b2  = (const float*)d_in[10];
    const float* enc_W  = (const float*)d_in[11];
    const float* enc_b  = (const float*)d_in[12];
    const float* dec_W  = (const float*)d_in[13];
    const float* dec_b  = (const float*)d_in[14];
    const float* sel_W1 = (const float*)d_in[15];
    const float* sel_b1 = (const float*)d_in[16];
    const float* sel_W2 = (const float*)d_in[17];
    const float* sel_b2 = (const float*)d_in[18];
    const int*   framec = (const int*)d_in[19];

    // workspace layout (bytes)
    char* ws = (char*)d_ws;
    const size_t SH  = (size_t)SEQ * HS * sizeof(float);   // 32 MB
    const size_t SC  = (size_t)SEQ * CS * sizeof(float);   // 16 MB
    size_t off = 0;
    int* flags = (int*)(ws + off); off += SEQ * sizeof(int);
    int* ksrc  = (int*)(ws + off); off += SEQ * sizeof(int);
    off = (off + 255) & ~(size_t)255;
    float* R0 = (float*)(ws + off); off += SC;   // selector hidden / enc output
    float* R1 = (float*)(ws + off); off += SH;   // prev  -> residual
    float* R3 = (float*)(ws + off); off += SH;   // A1    -> rec
    float* R5 = (float*)(ws + off); off += SH;   // pred
    (void)ws_size; (void)n_in; (void)in_sizes; (void)out_size;

    float* out_k = (float*)d_out;
    float* out_v = out_k + (size_t)SEQ * HS;

    const dim3 blk(256);
    const dim3 gN1024(CS / BN, SEQ / BM);   // N=1024 GEMMs
    const dim3 gN2048(HS / BN, SEQ / BM);   // N=2048 GEMMs
    const dim3 gElem(HS / 256, SEQ);
    const size_t nElem = (size_t)SEQ * HS;
    const dim3 gSub((unsigned)((nElem + 255) / 256));

    // ---- phase 1: selector over all timesteps ----
    gemm_f32_wmma<<<gN1024, blk, 0, stream>>>(keys, nullptr, sel_W1, sel_b1, nullptr,
                                              R0, SEQ, CS, HS, HS, 1);
    sel_score_kernel<<<dim3(SEQ), blk, 0, stream>>>(R0, sel_W2, sel_b2, framec, flags, CS);

    // ---- phase 2: keyframe scan ----
    kf_scan_kernel<<<1, 32, 0, stream>>>(flags, ksrc, SEQ);

    // ===================== K pipeline =====================
    gather_prev_kernel<<<gElem, blk, 0, stream>>>(keys, ksrc, R1, HS);
    // A1 = relu([keys | prev_k] @ km_W1 + b1)
    gemm_f32_wmma<<<gN2048, blk, 0, stream>>>(keys, R1, km_W1, km_b1, nullptr,
                                              R3, SEQ, HS, 2 * HS, HS, 1);
    // pred_k = prev_k + (A1 @ km_W2 + b2)
    gemm_f32_wmma<<<gN2048, blk, 0, stream>>>(R3, nullptr, km_W2, km_b2, R1,
                                              R5, SEQ, HS, HS, HS, 0);
    // rk = keys - pred_k
    sub_kernel<<<gSub, blk, 0, stream>>>(keys, R5, R1, nElem);
    // E = relu(rk @ enc_W + enc_b)
    gemm_f32_wmma<<<gN1024, blk, 0, stream>>>(R1, nullptr, enc_W, enc_b, nullptr,
                                              R0, SEQ, CS, HS, HS, 1);
    // rec_k = pred_k + relu(E @ dec_W + dec_b)
    gemm_f32_wmma<<<gN2048, blk, 0, stream>>>(R0, nullptr, dec_W, dec_b, R5,
                                              R3, SEQ, HS, CS, CS, 1);
    final_kernel<<<gElem, blk, 0, stream>>>(keys, R3, imp, flags, out_k, HS);

    // ===================== V pipeline (reuses R0/R1/R3/R5) =====================
    gather_prev_kernel<<<gElem, blk, 0, stream>>>(values, ksrc, R1, HS);
    gemm_f32_wmma<<<gN2048, blk, 0, stream>>>(values, R1, vm_W1, vm_b1, nullptr,
                                              R3, SEQ, HS, 2 * HS, HS, 1);
    gemm_f32_wmma<<<gN2048, blk, 0, stream>>>(R3, nullptr, vm_W2, vm_b2, R1,
                                              R5, SEQ, HS, HS, HS, 0);
    sub_kernel<<<gSub, blk, 0, stream>>>(values, R5, R1, nElem);
    gemm_f32_wmma<<<gN1024, blk, 0, stream>>>(R1, nullptr, enc_W, enc_b, nullptr,
                                              R0, SEQ, CS, HS, HS, 1);
    gemm_f32_wmma<<<gN2048, blk, 0, stream>>>(R0, nullptr, dec_W, dec_b, R5,
                                              R3, SEQ, HS, CS, CS, 1);
    final_kernel<<<gElem, blk, 0, stream>>>(values, R3, imp, flags, out_v, HS);
}